// VNDGCNN_45990509805765
// MI455X (gfx1250) — compile-verified
//
#include <hip/hip_runtime.h>

// ---------------------------------------------------------------------------
// VN-DGCNN edge-conv block for MI455X (gfx1250), wave32 + WMMA f32 16x16x4.
// Fused: Gram+top-k in LDS; BN stats inside layer GEMMs; layer1 gating inside
// layer2 tile build. Async global->LDS staging where the toolchain has it.
// ---------------------------------------------------------------------------

typedef __attribute__((ext_vector_type(2))) float v2f;
typedef __attribute__((ext_vector_type(8))) float v8f;

#define B_     8
#define N_     2048
#define CO_    64
#define KNN_   20
#define DF_    96                 // 3 * C
#define M1_    327680             // B_*N_*KNN_ columns per vector component
#define EPS_   1e-6f
#define BNEPS_ 1e-5f
#define DSTR   2052               // padded LDS stride for dist rows (banks)
#define BSTR   20                 // padded LDS stride for 16-wide K chunks

#if __has_builtin(__builtin_amdgcn_global_load_async_to_lds_b32) && \
    __has_builtin(__builtin_amdgcn_s_wait_asynccnt)
#define USE_ASYNC 1
#else
#define USE_ASYNC 0
#endif

static __device__ __forceinline__ v8f wmma4(v2f a, v2f b, v8f c) {
  // V_WMMA_F32_16X16X4_F32 : D(16x16,f32) = A(16x4,f32) * B(4x16,f32) + C
  return __builtin_amdgcn_wmma_f32_16x16x4_f32(false, a, false, b, (short)0, c,
                                               false, false);
}

#if USE_ASYNC
typedef __attribute__((address_space(1))) int as1_int;
typedef __attribute__((address_space(3))) int as3_int;
static __device__ __forceinline__ void async_cp_f32(const float* g, float* l) {
  __builtin_amdgcn_global_load_async_to_lds_b32(
      (as1_int*)(float*)g, (as3_int*)l, 0, 0);
}
#endif

// VN leaky-relu gating (BN folded into scale/shift).
static __device__ __forceinline__ void vn_gate(float p0, float p1, float p2,
                                               float d0, float d1, float d2,
                                               float sc, float sh,
                                               float& o0, float& o1, float& o2) {
  float norm = sqrtf(p0 * p0 + p1 * p1 + p2 * p2) + EPS_;
  float f = (norm * sc + sh) / norm;
  p0 *= f; p1 *= f; p2 *= f;
  float dot = p0 * d0 + p1 * d1 + p2 * d2;
  if (dot < 0.f) {
    float t = 0.8f * dot / (d0 * d0 + d1 * d1 + d2 * d2 + EPS_);
    o0 = p0 - t * d0; o1 = p1 - t * d1; o2 = p2 - t * d2;
  } else {
    o0 = p0; o1 = p1; o2 = p2;
  }
}

// --------------------------- prep kernels ----------------------------------

__global__ void k_zero(float* s) {
  int t = blockIdx.x * 256 + threadIdx.x;
  if (t < 512) s[t] = 0.f;
}

// x (B,32,3,N) -> xt (B,N,96) point-major (96 = c*3+v, knn feature order)
__global__ void k_transpose(const float* __restrict__ x, float* __restrict__ xt) {
  int tl = blockIdx.x * 256 + threadIdx.x;      // B*96*N threads
  int n  = tl & (N_ - 1);
  int cv = (tl >> 11) % DF_;
  int b  = tl / (N_ * DF_);
  xt[((size_t)(b * N_ + n)) * DF_ + cv] = x[((size_t)(b * DF_ + cv)) * N_ + n];
}

__global__ void k_xx(const float* __restrict__ xt, float* __restrict__ xx) {
  int tl = blockIdx.x * 256 + threadIdx.x;      // B*N threads
  const float* p = xt + (size_t)tl * DF_;
  float s = 0.f;
#pragma unroll 8
  for (int i = 0; i < DF_; ++i) s += p[i] * p[i];
  xx[tl] = s;
}

// ------------------ fused Gram (WMMA) + top-k, all in LDS -------------------
// Block = 16 rows x full 2048 cols of neg_dist (minus the row-constant xx_i,
// which does not change per-row ordering). Then wave-per-row top-20 select.
__global__ void __launch_bounds__(256) k_gramtopk(const float* __restrict__ xt,
                                                  const float* __restrict__ xx,
                                                  int* __restrict__ idxb) {
  __shared__ float sDist[16 * DSTR];            // 131,328 B
  __shared__ float sA[16 * BSTR];               // 16 rows x 16-k chunk
  __shared__ float sB[128 * BSTR];              // 128 cols x 16-k chunk
  const int t = threadIdx.x, l = t & 31, w = t >> 5;
  const int b = blockIdx.y, i0 = blockIdx.x * 16;
  const size_t xb = (size_t)b * N_ * DF_;

  for (int cb = 0; cb < N_; cb += 128) {
    v8f acc = {0, 0, 0, 0, 0, 0, 0, 0};
    for (int k0 = 0; k0 < DF_; k0 += 16) {
      __syncthreads();
      if (t < 64) {                             // stage A chunk (16x16)
        int r = t >> 2, c4 = (t & 3) << 2;
        float4 av = *(const float4*)(xt + xb + (size_t)(i0 + r) * DF_ + k0 + c4);
        *(float4*)&sA[r * BSTR + c4] = av;
      }
      // stage B chunk (128 cols x 16 k) : 2048 floats, 8 per thread
#pragma unroll
      for (int rep = 0; rep < 8; ++rep) {
        int e = t + rep * 256;
        int r = e >> 4, c = e & 15;
        const float* g = xt + xb + (size_t)(cb + r) * DF_ + k0 + c;
#if USE_ASYNC
        async_cp_f32(g, &sB[r * BSTR + c]);
#else
        sB[r * BSTR + c] = *g;
#endif
      }
#if USE_ASYNC
      __builtin_amdgcn_s_wait_asynccnt(0);
#endif
      __syncthreads();
#pragma unroll
      for (int kb = 0; kb < 16; kb += 4) {
        int kk = kb + ((l >> 4) << 1);
        v2f a, f;
        a.x = sA[(l & 15) * BSTR + kk];
        a.y = sA[(l & 15) * BSTR + kk + 1];
        f.x = sB[(w * 16 + (l & 15)) * BSTR + kk];
        f.y = sB[(w * 16 + (l & 15)) * BSTR + kk + 1];
        acc = wmma4(a, f, acc);
      }
    }
    // tile -> LDS dist rows (value = 2*inner - xx_col ; row term omitted)
    const int col = cb + w * 16 + (l & 15);
    const float xj = xx[b * N_ + col];
    const int r0 = (l >> 4) << 3;
#pragma unroll
    for (int r = 0; r < 8; ++r)
      sDist[(r0 + r) * DSTR + col] = 2.f * acc[r] - xj;
  }
  __syncthreads();

  // top-20 per row; wave handles rows w and w+8 (lane-private slots + shfl)
  for (int rr = 0; rr < 2; ++rr) {
    const int row = w + rr * 8;
    float* rv = sDist + row * DSTR;
    const int grow = b * N_ + i0 + row;
    float bv = -INFINITY; int bj = 1 << 30;
    for (int s = 0; s < 64; ++s) {
      int j = l + 32 * s; float vv = rv[j];
      if (vv > bv || (vv == bv && j < bj)) { bv = vv; bj = j; }
    }
    for (int sel = 0; sel < KNN_; ++sel) {
      float gv = bv; int gj = bj;
#pragma unroll
      for (int off = 16; off; off >>= 1) {
        float ov = __shfl_xor(gv, off, 32);
        int   oj = __shfl_xor(gj, off, 32);
        if (ov > gv || (ov == gv && oj < gj)) { gv = ov; gj = oj; }
      }
      if (l == 0) idxb[grow * KNN_ + sel] = gj;
      if ((gj & 31) == l) {                   // winner retires + rescans
        rv[gj] = -INFINITY;
        bv = -INFINITY; bj = 1 << 30;
        for (int s = 0; s < 64; ++s) {
          int j = l + 32 * s; float vv = rv[j];
          if (vv > bv || (vv == bv && j < bj)) { bv = vv; bj = j; }
        }
      }
    }
  }
}

// --------------- layer 1: gather + dual WMMA GEMM + fused stats -------------
// All 3 vector components per block; 8 waves = 4 o-blocks x {W->p, D->d}.
__global__ void __launch_bounds__(256) k_layer1f(const float* __restrict__ xt,
                                                 const int* __restrict__ idxb,
                                                 const float* __restrict__ W1,
                                                 const float* __restrict__ D1,
                                                 float* __restrict__ outP,
                                                 float* __restrict__ outD,
                                                 float* __restrict__ raw) {
  __shared__ float sW[64 * 65];
  __shared__ float sD[64 * 65];
  __shared__ float sE[3][64 * 17];
  const int t = threadIdx.x, l = t & 31, w = t >> 5;
  for (int e = t; e < 4096; e += 256) {
    int o = e >> 6, i = e & 63;
    sW[o * 65 + i] = W1[e];
    sD[o * 65 + i] = D1[e];
  }
  const int ob = (w & 3) << 4;
  const bool isP = (w < 4);
  const float* Wsel = isP ? sW : sD;
  float* outp = isP ? outP : outD;
  float st_s[8], st_q[8];
#pragma unroll
  for (int r = 0; r < 8; ++r) { st_s[r] = 0.f; st_q[r] = 0.f; }

  for (int mt = 0; mt < 4; ++mt) {
    const int m0 = (blockIdx.x * 4 + mt) << 4;
    __syncthreads();
    {   // gather edge tile for all 3 components
      int j = t & 15, i4 = (t >> 4) << 2;
      int m = m0 + j;
      int point = m / KNN_;
      int kk = m - point * KNN_;
      int nbr = idxb[point * KNN_ + kk];
      const float* cent = xt + (size_t)point * DF_;
      const float* nb   = xt + ((size_t)((point >> 11) << 11) + nbr) * DF_;
#pragma unroll
      for (int q = 0; q < 4; ++q) {
        int i = i4 + q;
        if (i < 32) {
          int base = i * 3;
#pragma unroll
          for (int v = 0; v < 3; ++v)
            sE[v][i * 17 + j] = nb[base + v] - cent[base + v];
        } else {
          int base = (i - 32) * 3;
#pragma unroll
          for (int v = 0; v < 3; ++v)
            sE[v][i * 17 + j] = cent[base + v];
        }
      }
    }
    __syncthreads();
    v8f acc[3];
#pragma unroll
    for (int v = 0; v < 3; ++v) {
      v8f z = {0, 0, 0, 0, 0, 0, 0, 0};
      acc[v] = z;
#pragma unroll
      for (int kb = 0; kb < 64; kb += 4) {
        int kk2 = kb + ((l >> 4) << 1);
        v2f a, bb;
        a.x  = Wsel[(ob + (l & 15)) * 65 + kk2];
        a.y  = Wsel[(ob + (l & 15)) * 65 + kk2 + 1];
        bb.x = sE[v][kk2 * 17 + (l & 15)];
        bb.y = sE[v][(kk2 + 1) * 17 + (l & 15)];
        acc[v] = wmma4(a, bb, acc[v]);
      }
    }
    const size_t colm = m0 + (l & 15);
    const int obase = ob + ((l >> 4) << 3);
#pragma unroll
    for (int v = 0; v < 3; ++v)
#pragma unroll
      for (int r = 0; r < 8; ++r)
        outp[((size_t)(v * CO_ + obase + r)) * M1_ + colm] = acc[v][r];
    if (isP) {   // fused BN statistics on p
#pragma unroll
      for (int r = 0; r < 8; ++r) {
        float a0 = acc[0][r], a1 = acc[1][r], a2 = acc[2][r];
        float nm = sqrtf(a0 * a0 + a1 * a1 + a2 * a2) + EPS_;
        st_s[r] += nm; st_q[r] += nm * nm;
      }
    }
  }
  if (isP) {
#pragma unroll
    for (int r = 0; r < 8; ++r) {
      float s = st_s[r], q = st_q[r];
#pragma unroll
      for (int off = 1; off <= 8; off <<= 1) {   // reduce within 16-lane halves
        s += __shfl_xor(s, off, 32);
        q += __shfl_xor(q, off, 32);
      }
      if ((l & 15) == 0) {
        int o = ob + r + ((l >> 4) << 3);
        atomicAdd(&raw[o * 2 + 0], s);
        atomicAdd(&raw[o * 2 + 1], q);
      }
    }
  }
}

// ------- layer 2: fused layer-1 gating in tile build + GEMM + stats ---------
// p2 written in place over d1 (block-local m range, barrier-ordered).
__global__ void __launch_bounds__(256) k_layer2f(const float* __restrict__ P1,
                                                 const float* D1b,
                                                 const float* __restrict__ fin1,
                                                 const float* __restrict__ W2,
                                                 const float* __restrict__ D2,
                                                 float* outP,
                                                 float* __restrict__ outD,
                                                 float* __restrict__ raw) {
  __shared__ float sW[64 * 65];
  __shared__ float sD[64 * 65];
  __shared__ float sE[3][64 * 17];
  __shared__ float sF[128];
  const int t = threadIdx.x, l = t & 31, w = t >> 5;
  for (int e = t; e < 4096; e += 256) {
    int o = e >> 6, i = e & 63;
    sW[o * 65 + i] = W2[e];
    sD[o * 65 + i] = D2[e];
  }
  if (t < 128) sF[t] = fin1[t];
  const int ob = (w & 3) << 4;
  const bool isP = (w < 4);
  const float* Wsel = isP ? sW : sD;
  float* outp = isP ? outP : outD;
  float st_s[8], st_q[8];
#pragma unroll
  for (int r = 0; r < 8; ++r) { st_s[r] = 0.f; st_q[r] = 0.f; }

  for (int mt = 0; mt < 4; ++mt) {
    const int m0 = (blockIdx.x * 4 + mt) << 4;
    __syncthreads();
    {   // build h1 tile: BN + VN-leaky-relu gating on the fly
      int j = t & 15, i4 = (t >> 4) << 2;
      size_t m = m0 + j;
#pragma unroll
      for (int q = 0; q < 4; ++q) {
        int i = i4 + q;
        size_t c0 = (size_t)i * M1_ + m;
        size_t c1 = c0 + (size_t)CO_ * M1_;
        size_t c2 = c1 + (size_t)CO_ * M1_;
        float h0, h1, h2;
        vn_gate(P1[c0], P1[c1], P1[c2], D1b[c0], D1b[c1], D1b[c2],
                sF[i * 2], sF[i * 2 + 1], h0, h1, h2);
        sE[0][i * 17 + j] = h0;
        sE[1][i * 17 + j] = h1;
        sE[2][i * 17 + j] = h2;
      }
    }
    __syncthreads();
    v8f acc[3];
#pragma unroll
    for (int v = 0; v < 3; ++v) {
      v8f z = {0, 0, 0, 0, 0, 0, 0, 0};
      acc[v] = z;
#pragma unroll
      for (int kb = 0; kb < 64; kb += 4) {
        int kk2 = kb + ((l >> 4) << 1);
        v2f a, bb;
        a.x  = Wsel[(ob + (l & 15)) * 65 + kk2];
        a.y  = Wsel[(ob + (l & 15)) * 65 + kk2 + 1];
        bb.x = sE[v][kk2 * 17 + (l & 15)];
        bb.y = sE[v][(kk2 + 1) * 17 + (l & 15)];
        acc[v] = wmma4(a, bb, acc[v]);
      }
    }
    const size_t colm = m0 + (l & 15);
    const int obase = ob + ((l >> 4) << 3);
#pragma unroll
    for (int v = 0; v < 3; ++v)
#pragma unroll
      for (int r = 0; r < 8; ++r)
        outp[((size_t)(v * CO_ + obase + r)) * M1_ + colm] = acc[v][r];
    if (isP) {
#pragma unroll
      for (int r = 0; r < 8; ++r) {
        float a0 = acc[0][r], a1 = acc[1][r], a2 = acc[2][r];
        float nm = sqrtf(a0 * a0 + a1 * a1 + a2 * a2) + EPS_;
        st_s[r] += nm; st_q[r] += nm * nm;
      }
    }
  }
  if (isP) {
#pragma unroll
    for (int r = 0; r < 8; ++r) {
      float s = st_s[r], q = st_q[r];
#pragma unroll
      for (int off = 1; off <= 8; off <<= 1) {
        s += __shfl_xor(s, off, 32);
        q += __shfl_xor(q, off, 32);
      }
      if ((l & 15) == 0) {
        int o = ob + r + ((l >> 4) << 3);
        atomicAdd(&raw[o * 2 + 0], s);
        atomicAdd(&raw[o * 2 + 1], q);
      }
    }
  }
}

// ------------------------- BN finalize + output ----------------------------

__global__ void k_fin(const float* __restrict__ raw, const float* __restrict__ g,
                      const float* __restrict__ be, float* __restrict__ fin) {
  int o = threadIdx.x;
  if (o < CO_) {
    const float ic = 1.0f / (float)M1_;
    float mean = raw[o * 2] * ic;
    float var  = raw[o * 2 + 1] * ic - mean * mean;
    float istd = rsqrtf(var + BNEPS_);
    float sc = istd * g[o];
    fin[o * 2 + 0] = sc;
    fin[o * 2 + 1] = be[o] - mean * sc;
  }
}

// final gate + mean-pool over k -> out (B, 64, 3, N)
__global__ void __launch_bounds__(256) k_final(const float* __restrict__ P,
                                               const float* __restrict__ Dm,
                                               const float* __restrict__ fin,
                                               float* __restrict__ out) {
  const int o  = blockIdx.y;
  const int pt = blockIdx.x * 256 + threadIdx.x;   // b*N+n
  const size_t m0 = (size_t)pt * KNN_;
  const size_t c0 = (size_t)o * M1_;
  const size_t c1 = c0 + (size_t)CO_ * M1_;
  const size_t c2 = c1 + (size_t)CO_ * M1_;
  float sc = fin[o * 2], sh = fin[o * 2 + 1];
  float a0 = 0.f, a1 = 0.f, a2 = 0.f;
#pragma unroll 4
  for (int kk = 0; kk < KNN_; ++kk) {
    size_t m = m0 + kk;
    float h0, h1, h2;
    vn_gate(P[c0 + m], P[c1 + m], P[c2 + m],
            Dm[c0 + m], Dm[c1 + m], Dm[c2 + m], sc, sh, h0, h1, h2);
    a0 += h0; a1 += h1; a2 += h2;
  }
  int b = pt >> 11, n = pt & (N_ - 1);
  size_t ob = ((size_t)(b * CO_ + o) * 3) * N_ + n;
  const float inv = 1.0f / (float)KNN_;
  out[ob]          = a0 * inv;
  out[ob + N_]     = a1 * inv;
  out[ob + 2 * N_] = a2 * inv;
}

// --------------------------- launcher --------------------------------------

extern "C" void kernel_launch(void* const* d_in, const int* in_sizes, int n_in,
                              void* d_out, int out_size, void* d_ws, size_t ws_size,
                              hipStream_t stream) {
  (void)in_sizes; (void)n_in; (void)out_size; (void)ws_size;
  const float* x  = (const float*)d_in[0];
  const float* W1 = (const float*)d_in[1];
  const float* D1 = (const float*)d_in[2];
  const float* g1 = (const float*)d_in[3];
  const float* b1 = (const float*)d_in[4];
  const float* W2 = (const float*)d_in[5];
  const float* D2 = (const float*)d_in[6];
  const float* g2 = (const float*)d_in[7];
  const float* b2 = (const float*)d_in[8];
  // d_in[9] = k (constant 20, hardcoded)

  constexpr size_t OFF_XT    = 0;                          // 6,291,456 B
  constexpr size_t OFF_XX    = 6291456;                    // 65,536 B
  constexpr size_t OFF_IDX   = 6356992;                    // 1,310,720 B
  constexpr size_t OFF_STATS = 7667712;                    // 4,096 B (512 f)
  constexpr size_t OFF_A     = 7671808;                    // 251,658,240 B
  constexpr size_t OFF_B     = OFF_A + 251658240;
  constexpr size_t OFF_C     = OFF_B + 251658240;
  char* ws = (char*)d_ws;
  float* xt    = (float*)(ws + OFF_XT);
  float* xx    = (float*)(ws + OFF_XX);
  int*   idxb  = (int*)  (ws + OFF_IDX);
  float* stats = (float*)(ws + OFF_STATS);
  float* raw1 = stats, *fin1 = stats + 128, *raw2 = stats + 256, *fin2 = stats + 384;
  float* bufA  = (float*)(ws + OFF_A);   // p1
  float* bufB  = (float*)(ws + OFF_B);   // d1 -> p2 (in place)
  float* bufC  = (float*)(ws + OFF_C);   // d2

  k_zero     <<<2, 256, 0, stream>>>(stats);
  k_transpose<<<6144, 256, 0, stream>>>(x, xt);
  k_xx       <<<64, 256, 0, stream>>>(xt, xx);
  k_gramtopk <<<dim3(128, 8), 256, 0, stream>>>(xt, xx, idxb);
  k_layer1f  <<<5120, 256, 0, stream>>>(xt, idxb, W1, D1, bufA, bufB, raw1);
  k_fin      <<<1, 64, 0, stream>>>(raw1, g1, b1, fin1);
  k_layer2f  <<<5120, 256, 0, stream>>>(bufA, bufB, fin1, W2, D2, bufB, bufC, raw2);
  k_fin      <<<1, 64, 0, stream>>>(raw2, g2, b2, fin2);
  k_final    <<<dim3(64, 64), 256, 0, stream>>>(bufB, bufC, fin2, (float*)d_out);
}